// PNNLayer_18210661335371
// MI455X (gfx1250) — compile-verified
//
#include <hip/hip_runtime.h>
#include <hip/hip_bf16.h>

#define N_NODES    10000
#define A_SETS     64
#define DLAT       128
#define CHUNKS     625      // N_NODES / 16
#define CHUNKS_PAD 640      // padded to a multiple of 16 for 16x16 M-tiles
#define INV_A      0.015625f

typedef __attribute__((ext_vector_type(2))) float v2f;
typedef __attribute__((ext_vector_type(8))) float v8f;

// ---------------------------------------------------------------------------
// K1: EC[t,:] = (1/64) * sum_{i=0..15} embeds[16t+i, :]   (pad rows 625..639 = 0)
// ---------------------------------------------------------------------------
__global__ void k_chunk_reduce(const float* __restrict__ embeds,
                               float* __restrict__ EC) {
  const int t = blockIdx.x;       // 0..639
  const int k = threadIdx.x;      // 0..127
  float s = 0.f;
  if (t < CHUNKS) {
    const float* p = embeds + (size_t)t * 16 * DLAT + k;
#pragma unroll
    for (int i = 0; i < 16; ++i) s += p[i * DLAT];
  }
  EC[t * DLAT + k] = s * INV_A;
}

// ---------------------------------------------------------------------------
// K2: C[640,128] = EC[640,128] @ W2^T   (W2 rows live in W_hidden[:,128:256],
// row stride 256 floats). One wave per 16x16 tile; K=128 in steps of 4.
// A-layout: lane L -> (M = L&15, K = kstep + 2*(L>>4) + {0,1})  (float2)
// B-layout: lane L -> (K = kstep + 2*(L>>4) + {0,1}, N = L&15)  (float2)
// D-layout: VGPR r -> (M = r + 8*(L>>4), N = L&15)
// ---------------------------------------------------------------------------
__global__ void k_gemm_C(const float* __restrict__ EC,
                         const float* __restrict__ W2,   // = W_hidden + 128
                         float* __restrict__ C) {
  const int lane = threadIdx.x & 31;
  const int wave = threadIdx.x >> 5;          // N tile 0..7
  const int mt   = blockIdx.x;                // M tile 0..39
  const int mrow = mt * 16 + (lane & 15);
  const int o    = wave * 16 + (lane & 15);
  const int kh   = (lane >> 4) * 2;
  const float* arow = EC + (size_t)mrow * DLAT;
  const float* brow = W2 + (size_t)o * (2 * DLAT);
  v8f acc = {};
#pragma unroll
  for (int k = 0; k < DLAT; k += 4) {
    v2f a = *(const v2f*)(arow + k + kh);
    v2f b = *(const v2f*)(brow + k + kh);
    acc = __builtin_amdgcn_wmma_f32_16x16x4_f32(false, a, false, b,
                                                (short)0, acc, false, false);
  }
  const int mbase = mt * 16 + (lane >> 4) * 8;
#pragma unroll
  for (int r = 0; r < 8; ++r) C[(mbase + r) * DLAT + o] = acc[r];
}

// ---------------------------------------------------------------------------
// K3: P[64,128] = (1/64) * embeds[ids] @ W1^T  (W1 = W_hidden[:, :128])
// ---------------------------------------------------------------------------
__global__ void k_gemm_P(const float* __restrict__ embeds,
                         const int* __restrict__ ids,
                         const float* __restrict__ W1,   // = W_hidden
                         float* __restrict__ P) {
  const int lane = threadIdx.x & 31;
  const int wave = threadIdx.x >> 5;          // N tile 0..7
  const int mt   = blockIdx.x;                // M tile 0..3
  const int m    = mt * 16 + (lane & 15);     // anchor index
  const int row  = ids[m];                    // gather
  const int o    = wave * 16 + (lane & 15);
  const int kh   = (lane >> 4) * 2;
  const float* arow = embeds + (size_t)row * DLAT;
  const float* brow = W1 + (size_t)o * (2 * DLAT);
  v8f acc = {};
#pragma unroll
  for (int k = 0; k < DLAT; k += 4) {
    v2f a = *(const v2f*)(arow + k + kh);
    v2f b = *(const v2f*)(brow + k + kh);
    acc = __builtin_amdgcn_wmma_f32_16x16x4_f32(false, a, false, b,
                                                (short)0, acc, false, false);
  }
  const int mbase = mt * 16 + (lane >> 4) * 8;
#pragma unroll
  for (int r = 0; r < 8; ++r) P[(mbase + r) * DLAT + o] = acc[r] * INV_A;
}

// ---------------------------------------------------------------------------
// K4: out[n,o] = b[o] + (dists^T @ P)[n,o] + sum_{i=0..3} C[(4n+i) % 625, o]
// A[m,k] = dists[k, n0+m] (column-strided gather, coalesced across lanes).
// ---------------------------------------------------------------------------
__global__ void k_out(const float* __restrict__ dists,
                      const float* __restrict__ P,
                      const float* __restrict__ C,
                      const float* __restrict__ bh,
                      float* __restrict__ out) {
  const int lane = threadIdx.x & 31;
  const int wave = threadIdx.x >> 5;          // N tile 0..7
  const int mt   = blockIdx.x;                // M tile 0..624
  const int na   = mt * 16 + (lane & 15);     // node index for A loads
  const int o    = wave * 16 + (lane & 15);
  const int kh   = (lane >> 4) * 2;
  v8f acc = {};
#pragma unroll
  for (int k = 0; k < A_SETS; k += 4) {
    v2f a;
    a.x = dists[(size_t)(k + kh)     * N_NODES + na];
    a.y = dists[(size_t)(k + kh + 1) * N_NODES + na];
    v2f b;
    b.x = P[(k + kh)     * DLAT + o];
    b.y = P[(k + kh + 1) * DLAT + o];
    acc = __builtin_amdgcn_wmma_f32_16x16x4_f32(false, a, false, b,
                                                (short)0, acc, false, false);
  }
  const float bias = bh[o];
  const int mbase = mt * 16 + (lane >> 4) * 8;
#pragma unroll
  for (int r = 0; r < 8; ++r) {
    const int n = mbase + r;
    int t0 = (4 * n) % CHUNKS;
    int t1 = t0 + 1; if (t1 >= CHUNKS) t1 -= CHUNKS;
    int t2 = t0 + 2; if (t2 >= CHUNKS) t2 -= CHUNKS;
    int t3 = t0 + 3; if (t3 >= CHUNKS) t3 -= CHUNKS;
    const float cs = C[t0 * DLAT + o] + C[t1 * DLAT + o] +
                     C[t2 * DLAT + o] + C[t3 * DLAT + o];
    out[(size_t)n * DLAT + o] = acc[r] + bias + cs;
  }
}

// ---------------------------------------------------------------------------
extern "C" void kernel_launch(void* const* d_in, const int* in_sizes, int n_in,
                              void* d_out, int out_size, void* d_ws, size_t ws_size,
                              hipStream_t stream) {
  const float* embeds = (const float*)d_in[0];   // [10000,128]
  const float* dists  = (const float*)d_in[1];   // [64,10000]
  const float* W      = (const float*)d_in[2];   // [128,256]
  const float* bh     = (const float*)d_in[3];   // [128]
  const int*   ids    = (const int*)d_in[4];     // [64]
  float* out = (float*)d_out;                    // [10000,128]

  char* ws = (char*)d_ws;
  float* EC = (float*)(ws);                                   // 640*128 f32
  float* C  = (float*)(ws + (size_t)CHUNKS_PAD * DLAT * 4);   // 640*128 f32
  float* P  = (float*)(ws + (size_t)2 * CHUNKS_PAD * DLAT * 4); // 64*128 f32

  k_chunk_reduce<<<CHUNKS_PAD, DLAT, 0, stream>>>(embeds, EC);
  k_gemm_C<<<CHUNKS_PAD / 16, 256, 0, stream>>>(EC, W + DLAT, C);
  k_gemm_P<<<A_SETS / 16, 256, 0, stream>>>(embeds, ids, W, P);
  k_out<<<N_NODES / 16, 256, 0, stream>>>(dists, P, C, bh, out);
}